// AttentionLayer_5540507811851
// MI455X (gfx1250) — compile-verified
//
#include <hip/hip_runtime.h>
#include <math.h>

// Problem constants (from reference): B=64, C1=1024, C2=2048, H=W=28
#define B_   64
#define C1_  1024
#define C2_  2048
#define P_   784     // H*W
#define NT   1024    // threads per attend block = 32 wave32s

typedef float v2f __attribute__((ext_vector_type(2)));
typedef float v8f __attribute__((ext_vector_type(8)));

// ---------------------------------------------------------------------------
// Kernel 1: q = input2 @ fc_w^T + fc_b   via V_WMMA_F32_16X16X4_F32 (fp32)
//   M = B_ (64), N = C1_ (1024), K = C2_ (2048)
//   A[m,k] = input2[m,k]   (row-major, contiguous in k)
//   B[k,n] = fc_w[n,k]     (fc_w row-major => contiguous in k per column n)
// ISA 32-bit A 16x4 layout: lane L: m = L%16, VGPR v holds K = v + 2*(L/16)
//   -> per lane a contiguous float2 at k0 + 2*(L/16). Same pairing for B.
// C/D 16x16 f32: lane L: n = L%16, VGPR r holds m = r + 8*(L/16).
// ---------------------------------------------------------------------------
__global__ __launch_bounds__(128) void fc_wmma_kernel(
    const float* __restrict__ in2,   // [B_,  C2_]
    const float* __restrict__ w,     // [C1_, C2_]
    const float* __restrict__ bias,  // [C1_]
    float* __restrict__ q)           // [B_,  C1_]
{
  const int wave  = threadIdx.x >> 5;
  const int lane  = threadIdx.x & 31;
  const int l16   = lane & 15;
  const int half  = lane >> 4;
  const int tileN = blockIdx.x * 4 + wave;   // 64 N-tiles total
  const int tileM = blockIdx.y;              // 4  M-tiles total

  const float* __restrict__ Arow = in2 + (size_t)(tileM * 16 + l16) * C2_;
  const float* __restrict__ Brow = w   + (size_t)(tileN * 16 + l16) * C2_;

  v8f acc0 = {};
  v8f acc1 = {};
  for (int k0 = 0; k0 < C2_; k0 += 8) {
    const int ka = k0 + 2 * half;
    v2f a0 = *(const v2f*)(Arow + ka);
    v2f b0 = *(const v2f*)(Brow + ka);
    v2f a1 = *(const v2f*)(Arow + ka + 4);
    v2f b1 = *(const v2f*)(Brow + ka + 4);
    // 8-arg form: (neg_a, A, neg_b, B, c_mod, C, reuse_a, reuse_b)
    acc0 = __builtin_amdgcn_wmma_f32_16x16x4_f32(false, a0, false, b0,
                                                 (short)0, acc0, false, false);
    acc1 = __builtin_amdgcn_wmma_f32_16x16x4_f32(false, a1, false, b1,
                                                 (short)0, acc1, false, false);
  }

  const int   n  = tileN * 16 + l16;
  const float bv = bias[n];
#pragma unroll
  for (int r = 0; r < 8; ++r) {
    const int m = tileM * 16 + r + 8 * half;
    q[(size_t)m * C1_ + n] = acc0[r] + acc1[r] + bv;
  }
}

// ---------------------------------------------------------------------------
// Kernel 2: fused scores -> softmax -> weighted spatial sum, one block per b.
// input1[b] is a 3.2 MB [C1_, P_] slab: both passes stay L2-resident.
// ---------------------------------------------------------------------------
__device__ __forceinline__ float wave_max(float v) {
#pragma unroll
  for (int m = 16; m > 0; m >>= 1) v = fmaxf(v, __shfl_xor(v, m, 32));
  return v;
}
__device__ __forceinline__ float wave_sum(float v) {
#pragma unroll
  for (int m = 16; m > 0; m >>= 1) v += __shfl_xor(v, m, 32);
  return v;
}

__global__ __launch_bounds__(NT) void attend_kernel(
    const float* __restrict__ x,   // [B_, C1_, P_]
    const float* __restrict__ q,   // [B_, C1_]
    float* __restrict__ out)       // [B_, C1_]
{
  __shared__ float s_q[C1_];       // 4 KB
  __shared__ float s_attn[P_];     // 3136 B
  __shared__ float s_red[32];
  __shared__ float s_max, s_sum;

  const int b    = blockIdx.x;
  const int tid  = threadIdx.x;
  const int wave = tid >> 5;
  const int lane = tid & 31;
  const float* __restrict__ xb = x + (size_t)b * C1_ * P_;

  // stage q[b,:] into LDS (broadcast reads in phase A)
  s_q[tid] = q[(size_t)b * C1_ + tid];
  __syncthreads();

  // ---- Phase A: scores[p] = sum_c x[c,p] * q[c]  (coalesced along p) ----
  const int p = tid;
  float score = 0.0f;
  if (p < P_) {
#pragma unroll 4
    for (int c = 0; c < C1_; ++c) {
      if ((c & 7) == 0 && (c + 16) < C1_)
        __builtin_prefetch(xb + (size_t)(c + 16) * P_ + p, 0, 1);
      score = fmaf(xb[(size_t)c * P_ + p], s_q[c], score);
    }
  }

  // ---- softmax over p (two-level wave32 reductions) ----
  float wm = wave_max((p < P_) ? score : -INFINITY);
  if (lane == 0) s_red[wave] = wm;
  __syncthreads();
  if (wave == 0) {
    float t = wave_max(s_red[lane]);
    if (lane == 0) s_max = t;
  }
  __syncthreads();
  const float mx = s_max;

  const float e = (p < P_) ? __expf(score - mx) : 0.0f;
  float ws = wave_sum(e);
  if (lane == 0) s_red[wave] = ws;
  __syncthreads();
  if (wave == 0) {
    float t = wave_sum(s_red[lane]);
    if (lane == 0) s_sum = t;
  }
  __syncthreads();
  const float inv = 1.0f / s_sum;
  if (p < P_) s_attn[p] = e * inv;
  __syncthreads();

  // ---- Phase C: out[b,c] = sum_p x[c,p] * attn[p] ----
  // wave-per-channel, lanes stride p by 32 -> 128B coalesced loads, L2 hits.
#pragma unroll 1
  for (int j = 0; j < C1_ / 32; ++j) {
    const int c = j * 32 + wave;
    const float* __restrict__ row = xb + (size_t)c * P_;
    float part = 0.0f;
#pragma unroll 5
    for (int i = 0; i < 25; ++i) {       // 25*32 = 800 >= 784
      const int pp = lane + 32 * i;
      if (pp < P_) part = fmaf(row[pp], s_attn[pp], part);
    }
    part = wave_sum(part);
    if (lane == 0) out[(size_t)b * C1_ + c] = part;
  }
}

// ---------------------------------------------------------------------------
// Host entry
// ---------------------------------------------------------------------------
extern "C" void kernel_launch(void* const* d_in, const int* in_sizes, int n_in,
                              void* d_out, int out_size, void* d_ws, size_t ws_size,
                              hipStream_t stream) {
  (void)in_sizes; (void)n_in; (void)out_size; (void)ws_size;

  const float* input1 = (const float*)d_in[0];  // [B, C1, H, W]
  const float* input2 = (const float*)d_in[1];  // [B, C2]
  const float* fc_w   = (const float*)d_in[2];  // [C1, C2]
  const float* fc_b   = (const float*)d_in[3];  // [C1]

  float* q   = (float*)d_ws;    // [B, C1] = 256 KB scratch
  float* out = (float*)d_out;   // [B, C1]

  dim3 g1(C1_ / 64, B_ / 16);   // (16, 4): 4 waves/block, one 16x16 tile each
  fc_wmma_kernel<<<g1, 128, 0, stream>>>(input2, fc_w, fc_b, q);

  attend_kernel<<<B_, NT, 0, stream>>>(input1, q, out);
}